// ScribblePooling_42760694399081
// MI455X (gfx1250) — compile-verified
//
#include <hip/hip_runtime.h>
#include <hip/hip_bf16.h>

// ---------------- problem constants (match reference) ----------------
#define B_   8
#define C_   256
#define LL   9216      // 96*96
#define KK   5         // label channels
#define ML   512
#define NTBL 32        // B_ * 4 slots (3 objects + background)

static const long long OFF_SRC_OBJ  = 0LL;        // 24*512*256
static const long long OFF_MASK_OBJ = 3145728LL;  // +24*512
static const long long OFF_SRC_BG   = 3158016LL;  // +8*512*256
static const long long OFF_MASK_BG  = 4206592LL;  // +8*512
static const long long OFF_SRC_OTH  = 4210688LL;  // +24*1536*256
static const long long OFF_MASK_OTH = 13647872LL; // total 13684736

typedef unsigned int u32x4 __attribute__((ext_vector_type(4)));
typedef int          i32x8 __attribute__((ext_vector_type(8)));
typedef int          i32x4 __attribute__((ext_vector_type(4)));

// ---------------- kernel 1: per-(b,slot) compaction scan ----------------
// grid = 32 blocks (b*4+slot), block = 256 threads (8 wave32s).
// The 36 KB contiguous label plane is DMA'd into LDS by the Tensor Data
// Mover (tensor_load_to_lds + s_wait_tensorcnt). A verified coalesced
// staging pass then (re)writes the identical data — correctness hedge for
// the uncharacterized builtin-arg -> D# mapping; it hits L2 if the DMA
// already landed the plane. Block-wide prefix scan reproduces the
// reference's cumsum-rank compaction exactly.
__global__ __launch_bounds__(256)
void scribble_scan_kernel(const int* __restrict__ label,
                          const int* __restrict__ num_objects,
                          int* __restrict__ sel,   // [NTBL][ML]
                          int* __restrict__ count) // [NTBL]
{
    __shared__ int lbl[LL];      // 36 KB of 320 KB/WGP LDS
    __shared__ int psum[256];

    const int tid = threadIdx.x;
    const int blk = blockIdx.x;
    const int b = blk >> 2, s = blk & 3;
    const int ch = (s < 3) ? (s + 1) : (KK - 1);
    const bool active = (s == 3) || (s < num_objects[b]);

    const int* __restrict__ src = label + ((long long)b * KK + ch) * LL;

#if __has_builtin(__builtin_amdgcn_tensor_load_to_lds)
    // ---- TDM: single DMA of a 9216 x 4B (1-line) tile, global -> LDS ----
    // D# groups 0/1 per CDNA5 ISA; trailing
    // groups zero (<=2D tensor => NULL-group mode).
    if (tid < 32) {   // TDM is a per-wave op (EXEC ignored); one wave issues it
        const unsigned lds_base = (unsigned)(uintptr_t)lbl;            // addr[31:0] = LDS offset
        const unsigned long long ga = (unsigned long long)(uintptr_t)src;
        const u32x4 g0 = { 1u,                                         // count=1, user mode
                           lds_base,                                   // lds_addr
                           (unsigned)(ga & 0xFFFFFFFFu),               // global_addr[31:0]
                           ((unsigned)((ga >> 32) & 0x1FFFFFFu))       // global_addr[56:32]
                             | 0x80000000u };                          // type=2 ("image")
        const i32x8 g1 = { 0x00020000,                                 // data_size=2 (4B), no mask/pad/iter
                           (int)(9216u << 16),                         // tensor_dim0[15:0] in [31:16]
                           0x00010000,                                 // tensor_dim0[31:16]=0 | tensor_dim1=1
                           (int)(9216u << 16),                         // tile_dim0 = 9216
                           0,                                          // tile_dim1=0, tile_dim2=0 (unused)
                           9216,                                       // tensor_dim0_stride[31:0]
                           0, 0 };                                     // stride hi / dim1_stride = 0
        const i32x4 gz4 = { 0, 0, 0, 0 };
        const i32x8 gz8 = { 0, 0, 0, 0, 0, 0, 0, 0 };
        __builtin_amdgcn_tensor_load_to_lds(g0, g1, gz4, gz4, gz8, 0); // clang-23 6-arg form
#if __has_builtin(__builtin_amdgcn_s_wait_tensorcnt)
        __builtin_amdgcn_s_wait_tensorcnt(0);
#else
        asm volatile("s_wait_tensorcnt 0x0" ::: "memory");
#endif
    }
    __syncthreads();   // DMA complete & visible before the verified staging pass
#endif

    // ---- verified coalesced staging (identical data; L2-hot if TDM landed) ----
#pragma unroll
    for (int i = 0; i < LL / 256; ++i)
        lbl[i * 256 + tid] = src[i * 256 + tid];
    __syncthreads();

    const int base = tid * (LL / 256);
    int cnt = 0;
    if (active) {
#pragma unroll
        for (int j = 0; j < LL / 256; ++j) cnt += (lbl[base + j] == 1);
    }
    psum[tid] = cnt;
    __syncthreads();
    // Hillis-Steele inclusive scan over 256 partials
    for (int off = 1; off < 256; off <<= 1) {
        int v = (tid >= off) ? psum[tid - off] : 0;
        __syncthreads();
        psum[tid] += v;
        __syncthreads();
    }
    const int excl  = psum[tid] - cnt;
    const int total = psum[255];

    if (active) {
        int r = excl;
#pragma unroll 4
        for (int j = 0; j < LL / 256; ++j) {
            if (lbl[base + j] == 1) {
                if (r < ML) sel[blk * ML + r] = base + j;
                ++r;
            }
        }
    }
    if (tid == 0) count[blk] = active ? min(total, ML) : 0;
}

// ---------------- row decode for the unified gather ----------------
// Global output rows: [0,12288) src_obj | [12288,16384) src_bg | [16384,53248) src_other
__device__ __forceinline__
void decode_row(long long row, int& b, int& tbl, int& r, long long& dst, long long& mdst)
{
    if (row < 12288) {                       // src_obj: bo = b*3+o
        int bo = (int)(row >> 9); r = (int)(row & 511);
        b = bo / 3; int o = bo - 3 * b;
        tbl = (b << 2) + o;
        dst  = OFF_SRC_OBJ  + row * C_;
        mdst = OFF_MASK_OBJ + row;
    } else if (row < 16384) {                // src_bg
        long long i = row - 12288;
        b = (int)(i >> 9); r = (int)(i & 511);
        tbl = (b << 2) + 3;
        dst  = OFF_SRC_BG  + i * C_;
        mdst = OFF_MASK_BG + i;
    } else {                                 // src_other: idx[o] = {j!=o in 0..2} ++ {bg}
        long long i = row - 16384;
        int bo = (int)(i / 1536); int q = (int)(i - (long long)bo * 1536);
        int jj = q >> 9; r = q & 511;
        b = bo / 3; int o = bo - 3 * b;
        int slot = (jj == 2) ? 3 : ((jj >= o) ? jj + 1 : jj);
        tbl = (b << 2) + slot;
        dst  = OFF_SRC_OTH  + i * C_;
        mdst = OFF_MASK_OTH + i;
    }
}

// ---------------- kernel 2: unified transposing gather ----------------
// block = 256 threads = one full channel vector (coalesced 1 KB store per row);
// each block handles 8 rows: decode + global_prefetch all scattered sources,
// then load/store. feats (75.5 MB) is L2-resident (192 MB L2), so the
// stride-36KB per-lane gather is served at L2 bandwidth.
#define ROWS_PER_BLK 8
__global__ __launch_bounds__(256)
void scribble_gather_kernel(const float* __restrict__ feats,
                            const int*  __restrict__ sel,
                            const int*  __restrict__ count,
                            float* __restrict__ out)
{
    const int c = threadIdx.x;
    const long long row0 = (long long)blockIdx.x * ROWS_PER_BLK;

    const float* srcp[ROWS_PER_BLK];
    long long dstk[ROWS_PER_BLK], mdstk[ROWS_PER_BLK];
    bool valid[ROWS_PER_BLK];

#pragma unroll
    for (int k = 0; k < ROWS_PER_BLK; ++k) {
        int b, tbl, r; long long dst, mdst;
        decode_row(row0 + k, b, tbl, r, dst, mdst);
        const int cn = count[tbl];               // uniform -> scalar load
        valid[k] = (r < cn);
        dstk[k] = dst; mdstk[k] = mdst;
        const float* p = feats;                  // dummy init
        if (valid[k]) {
            const int l = sel[tbl * ML + r];
            p = feats + ((long long)b * C_ + c) * LL + l;
            __builtin_prefetch(p, 0, 3);         // -> global_prefetch_b8 (no LOADcnt)
        }
        srcp[k] = p;
    }

#pragma unroll
    for (int k = 0; k < ROWS_PER_BLK; ++k) {
        const float v = valid[k] ? *srcp[k] : 0.0f;
        out[dstk[k] + c] = v;
        if (c == 0) out[mdstk[k]] = valid[k] ? 0.0f : 1.0f;  // pad mask
    }
}

// ---------------- launch ----------------
extern "C" void kernel_launch(void* const* d_in, const int* in_sizes, int n_in,
                              void* d_out, int out_size, void* d_ws, size_t ws_size,
                              hipStream_t stream) {
    const float* feats       = (const float*)d_in[0];  // [8,256,96,96] f32
    const int*   label       = (const int*)d_in[1];    // [8,5,96,96] i32
    const int*   num_objects = (const int*)d_in[2];    // [8] i32
    float* out = (float*)d_out;

    // workspace: 32 counts + 32*512 selected-pixel indices (65.7 KB)
    int* ws_count = (int*)d_ws;
    int* ws_sel   = ws_count + NTBL;

    scribble_scan_kernel<<<NTBL, 256, 0, stream>>>(label, num_objects, ws_sel, ws_count);

    const long long total_rows = 12288 + 4096 + 36864;        // 53248
    const int nblk = (int)(total_rows / ROWS_PER_BLK);        // 6656
    scribble_gather_kernel<<<nblk, 256, 0, stream>>>(feats, ws_sel, ws_count, out);
}